// SimpleFunctionRecurrentNetwork_38354057953441
// MI455X (gfx1250) — compile-verified
//
#include <hip/hip_runtime.h>

typedef __attribute__((ext_vector_type(2))) float v2f;
typedef __attribute__((ext_vector_type(4))) float v4f;
typedef __attribute__((ext_vector_type(8))) float v8f;

#define DEV __device__ __forceinline__

// CDNA5 hardware tanh (TRANS op). Per ISA 7.4, a trans op needs one
// independent op / V_NOP before its result is consumed; since this is inline
// asm the compiler may not know, so bundle a v_nop.
DEV float htanh(float x) {
  float r;
  asm("v_tanh_f32 %0, %1\n\tv_nop" : "=v"(r) : "v"(x));
  return r;
}
// sigmoid(x) = 0.5 * tanh(x/2) + 0.5  -> one trans instruction
DEV float hsigm(float x) { return __builtin_fmaf(0.5f, htanh(0.5f * x), 0.5f); }

// ---------------------------------------------------------------------------
// Phase 1: xg[row, g] = sum_d x[row, d] * W_ih[g, d] + (b_ih[g] + b_hh[g])
// via V_WMMA_F32_16X16X4_F32.  row = b*T + t (rows contiguous in memory).
// A: 16x4 f32 slice of x rows. B: 4x16 = W_ih^T slice (cols >= 8 zero).
// C: bias broadcast. K = 10 padded to 12 (3 WMMAs), pad entries zeroed.
// ---------------------------------------------------------------------------
__global__ void xg_wmma_kernel(const float* __restrict__ x,
                               const float* __restrict__ W_ih,
                               const float* __restrict__ b_ih,
                               const float* __restrict__ b_hh,
                               float* __restrict__ xg,
                               long numTiles) {
  const int lane = threadIdx.x & 31;
  const int n    = lane & 15;   // N column (gate index; valid if < 8)
  const int half = lane >> 4;   // K-half selector for A/B, M-half for C/D

  // B-matrix slices: B[k][n] = W_ih[n*10 + k] (n<8, k<10), layout k = v + 2*half
  v2f B0, B1, B2;
  {
    int k0 = 2 * half;
    B0.x = (n < 8) ? W_ih[n * 10 + k0 + 0] : 0.0f;
    B0.y = (n < 8) ? W_ih[n * 10 + k0 + 1] : 0.0f;
    B1.x = (n < 8) ? W_ih[n * 10 + k0 + 4] : 0.0f;
    B1.y = (n < 8) ? W_ih[n * 10 + k0 + 5] : 0.0f;
    int k2 = 8 + k0;
    B2.x = (n < 8 && k2 + 0 < 10) ? W_ih[n * 10 + k2 + 0] : 0.0f;
    B2.y = (n < 8 && k2 + 1 < 10) ? W_ih[n * 10 + k2 + 1] : 0.0f;
  }
  const float bias = (n < 8) ? (b_ih[n] + b_hh[n]) : 0.0f;
  v8f cbias;
#pragma unroll
  for (int i = 0; i < 8; ++i) cbias[i] = bias;

  const long wavesTotal = (long)gridDim.x * (blockDim.x >> 5);
  const long wid = (long)blockIdx.x * (blockDim.x >> 5) + (threadIdx.x >> 5);

  for (long tile = wid; tile < numTiles; tile += wavesTotal) {
    const float* xr = x + (tile * 16 + n) * 10;  // A row for this lane (M = n)
    // A layout: vgpr v holds k = v + 2*half; slices at k0 = 0, 4, 8
    v2f a0 = __builtin_nontemporal_load((const v2f*)(xr + 2 * half));
    v2f a1 = __builtin_nontemporal_load((const v2f*)(xr + 4 + 2 * half));
    // k = 10,11 (half==1) must be zero; load valid k=8,9 then branchless select
    v2f a2t = __builtin_nontemporal_load((const v2f*)(xr + 8));
    v2f zero = {0.0f, 0.0f};
    v2f a2 = half ? zero : a2t;

    v8f acc = __builtin_amdgcn_wmma_f32_16x16x4_f32(false, a0, false, B0,
                                                    (short)0, cbias, false, false);
    acc = __builtin_amdgcn_wmma_f32_16x16x4_f32(false, a1, false, B1,
                                                (short)0, acc, false, false);
    acc = __builtin_amdgcn_wmma_f32_16x16x4_f32(false, a2, false, B2,
                                                (short)0, acc, false, false);

    // D layout: component v holds row M = v + 8*half, column n. Keep n<8.
    if (n < 8) {
      float* orow = xg + (tile * 16 + half * 8) * 8 + n;
#pragma unroll
      for (int v = 0; v < 8; ++v) orow[v * 8] = acc[v];
    }
  }
}

// ---------------------------------------------------------------------------
// Phase 2: sequential LSTM scan, one thread per batch element.
// Gate order (torch LSTMCell): i, f, g, o in chunks of H=2.
// Double-buffer next timestep's 32B of xg to hide load latency.
// ---------------------------------------------------------------------------
__global__ void lstm_scan_kernel(const float* __restrict__ xg,
                                 const float* __restrict__ h0v,
                                 const float* __restrict__ c0v,
                                 const float* __restrict__ W_hh,
                                 const float* __restrict__ W_out,
                                 const float* __restrict__ b_out,
                                 float* __restrict__ out, int T, int B) {
  int b = blockIdx.x * blockDim.x + threadIdx.x;
  if (b >= B) return;

  float w[16];
#pragma unroll
  for (int i = 0; i < 16; ++i) w[i] = W_hh[i];

  float h0 = h0v[0], h1 = h0v[1];
  float c0 = c0v[0], c1 = c0v[1];

  const v4f* p = (const v4f*)(xg + (long)b * T * 8);
  v4f ga = __builtin_nontemporal_load(p);
  v4f gb = __builtin_nontemporal_load(p + 1);

  for (int t = 0; t < T; ++t) {
    v4f na = ga, nb = gb;
    if (t + 1 < T) {
      na = __builtin_nontemporal_load(p + 2);
      nb = __builtin_nontemporal_load(p + 3);
    }
    float i0 = hsigm(ga.x + w[0] * h0 + w[1] * h1);
    float i1 = hsigm(ga.y + w[2] * h0 + w[3] * h1);
    float f0 = hsigm(ga.z + w[4] * h0 + w[5] * h1);
    float f1 = hsigm(ga.w + w[6] * h0 + w[7] * h1);
    float g0 = htanh(gb.x + w[8] * h0 + w[9] * h1);
    float g1 = htanh(gb.y + w[10] * h0 + w[11] * h1);
    float o0 = hsigm(gb.z + w[12] * h0 + w[13] * h1);
    float o1 = hsigm(gb.w + w[14] * h0 + w[15] * h1);
    c0 = f0 * c0 + i0 * g0;
    c1 = f1 * c1 + i1 * g1;
    h0 = o0 * htanh(c0);
    h1 = o1 * htanh(c1);
    ga = na;
    gb = nb;
    p += 2;
  }
  out[b] = h0 * W_out[0] + h1 * W_out[1] + b_out[0];
}

// ---------------------------------------------------------------------------
// Fallback: fully fused (no workspace needed). One thread per batch element,
// 80 FMA/step for the x projection. Used if ws_size is too small or B*T is not a multiple of 16.
// ---------------------------------------------------------------------------
__global__ void lstm_fused_kernel(const float* __restrict__ x,
                                  const float* __restrict__ h0v,
                                  const float* __restrict__ c0v,
                                  const float* __restrict__ W_ih,
                                  const float* __restrict__ W_hh,
                                  const float* __restrict__ b_ih,
                                  const float* __restrict__ b_hh,
                                  const float* __restrict__ W_out,
                                  const float* __restrict__ b_out,
                                  float* __restrict__ out, int T, int B) {
  int b = blockIdx.x * blockDim.x + threadIdx.x;
  if (b >= B) return;
  float wih[80], w[16], bias[8];
#pragma unroll
  for (int i = 0; i < 80; ++i) wih[i] = W_ih[i];
#pragma unroll
  for (int i = 0; i < 16; ++i) w[i] = W_hh[i];
#pragma unroll
  for (int i = 0; i < 8; ++i) bias[i] = b_ih[i] + b_hh[i];
  float h0 = h0v[0], h1 = h0v[1], c0 = c0v[0], c1 = c0v[1];
  const float* xr = x + (long)b * T * 10;
  for (int t = 0; t < T; ++t, xr += 10) {
    float xv[10];
#pragma unroll
    for (int d = 0; d < 5; ++d) {
      v2f v = __builtin_nontemporal_load((const v2f*)(xr + 2 * d));
      xv[2 * d] = v.x;
      xv[2 * d + 1] = v.y;
    }
    float g[8];
#pragma unroll
    for (int j = 0; j < 8; ++j) {
      float a = bias[j] + w[2 * j] * h0 + w[2 * j + 1] * h1;
#pragma unroll
      for (int d = 0; d < 10; ++d) a += xv[d] * wih[j * 10 + d];
      g[j] = a;
    }
    float i0 = hsigm(g[0]), i1 = hsigm(g[1]), f0 = hsigm(g[2]), f1 = hsigm(g[3]);
    float t0 = htanh(g[4]), t1 = htanh(g[5]), o0 = hsigm(g[6]), o1 = hsigm(g[7]);
    c0 = f0 * c0 + i0 * t0;
    c1 = f1 * c1 + i1 * t1;
    h0 = o0 * htanh(c0);
    h1 = o1 * htanh(c1);
  }
  out[b] = h0 * W_out[0] + h1 * W_out[1] + b_out[0];
}

extern "C" void kernel_launch(void* const* d_in, const int* in_sizes, int n_in,
                              void* d_out, int out_size, void* d_ws, size_t ws_size,
                              hipStream_t stream) {
  (void)n_in;
  const float* x     = (const float*)d_in[0];
  const float* h0    = (const float*)d_in[1];
  const float* c0    = (const float*)d_in[2];
  const float* W_ih  = (const float*)d_in[3];
  const float* W_hh  = (const float*)d_in[4];
  const float* b_ih  = (const float*)d_in[5];
  const float* b_hh  = (const float*)d_in[6];
  const float* W_out = (const float*)d_in[7];
  const float* b_out = (const float*)d_in[8];
  float* out = (float*)d_out;

  const int D = 10;
  const int B = out_size;                    // output is [B,1]
  const long totalX = (long)in_sizes[0];     // B*T*D
  const int T = (int)(totalX / ((long)B * D));
  const long nrows = (long)B * T;
  const size_t need = (size_t)nrows * 8 * sizeof(float);

  if (ws_size >= need && (nrows % 16) == 0) {
    float* xg = (float*)d_ws;
    const long numTiles = nrows / 16;
    int nBlocks = 512;  // 4096 waves, grid-stride over 256k tiles
    if ((long)nBlocks * 8 > numTiles) nBlocks = (int)((numTiles + 7) / 8);
    if (nBlocks < 1) nBlocks = 1;
    xg_wmma_kernel<<<nBlocks, 256, 0, stream>>>(x, W_ih, b_ih, b_hh, xg, numTiles);
    const int sb = 128;
    lstm_scan_kernel<<<(B + sb - 1) / sb, sb, 0, stream>>>(xg, h0, c0, W_hh,
                                                           W_out, b_out, out, T, B);
  } else {
    const int sb = 128;
    lstm_fused_kernel<<<(B + sb - 1) / sb, sb, 0, stream>>>(
        x, h0, c0, W_ih, W_hh, b_ih, b_hh, W_out, b_out, out, T, B);
  }
}